// GCN1_22514218566432
// MI455X (gfx1250) — compile-verified
//
#include <hip/hip_runtime.h>
#include <hip/hip_bf16.h>

typedef float v2f __attribute__((ext_vector_type(2)));
typedef float v8f __attribute__((ext_vector_type(8)));

#define NGRAPH 128
#define FHEAD  32
#define MT     5      // M-tiles per wave in the WMMA GEMM (3125 = 5*625, no tail)

// ---------------- utility kernels ----------------

__global__ void k_zero(float* __restrict__ p, size_t n) {
    size_t i = (size_t)blockIdx.x * blockDim.x + threadIdx.x;
    if (i < n) p[i] = 0.0f;
}

__global__ void k_negcopy(float* __restrict__ d, const float* __restrict__ s, size_t n) {
    size_t i = (size_t)blockIdx.x * blockDim.x + threadIdx.x;
    if (i < n) d[i] = -s[i];
}

// deg[src[e]] += attr[e]
__global__ void k_degree(const int* __restrict__ src, const float* __restrict__ attr,
                         float* __restrict__ deg, int E) {
    int e = blockIdx.x * blockDim.x + threadIdx.x;
    if (e < E) atomicAdd(&deg[src[e]], attr[e]);
}

// deg -> dinv (in place)
__global__ void k_dinv(float* __restrict__ deg, int n) {
    int i = blockIdx.x * blockDim.x + threadIdx.x;
    if (i < n) {
        float d = deg[i];
        deg[i] = (d > 0.0f) ? __frsqrt_rn(d) : 0.0f;
    }
}

// w[e] = -dinv[src]*attr*dinv[dst]
__global__ void k_edgew(const int* __restrict__ src, const int* __restrict__ dst,
                        const float* __restrict__ attr, const float* __restrict__ dinv,
                        float* __restrict__ w, int E) {
    int e = blockIdx.x * blockDim.x + threadIdx.x;
    if (e < E) w[e] = -dinv[src[e]] * attr[e] * dinv[dst[e]];
}

// out[dst[e], f0..f0+3] += scale * w[e] * xin[src[e], f0..f0+3]
// 2D grid: x over edges (coalesced w/src/dst), y over feature chunks of 4.
__global__ void k_prop(const float* __restrict__ w, const int* __restrict__ src,
                       const int* __restrict__ dst, const float* __restrict__ xin,
                       float* __restrict__ xout, int E, int F, float scale) {
    int e = blockIdx.x * blockDim.x + threadIdx.x;
    if (e >= E) return;
    int f0 = blockIdx.y * 4;
    float c = scale * w[e];
    float4 v = *(const float4*)(xin + (size_t)src[e] * F + f0);
    float* o = xout + (size_t)dst[e] * F + f0;
    atomicAdd(o + 0, c * v.x);
    atomicAdd(o + 1, c * v.y);
    atomicAdd(o + 2, c * v.z);
    atomicAdd(o + 3, c * v.w);
}

// ---------------- fused Chebyshev GEMM: out = relu(T0 W0 + T1 W1 + T2 W2 + b) ----------------
// One wave computes MT vertical 16x16 tiles of one 16-wide output column.
// V_WMMA_F32_16X16X4_F32; B fragment loaded once per K-step, reused by MT WMMAs.
// Tail handling is done by clamping the tile base (overlapping waves recompute
// identical values), so there is a single fully-unrolled register-resident path.
// A (16x4 f32): lanes 0-15 hold K={k,k+1}, lanes 16-31 hold K={k+2,k+3} (ISA 7.12.2).
// C/D (16x16 f32): VGPR j = rows j (lanes 0-15) and j+8 (lanes 16-31).
__global__ __launch_bounds__(32) void k_cheb_gemm3(
    const float* __restrict__ T0, const float* __restrict__ T1, const float* __restrict__ T2,
    const float* __restrict__ W,    // [3][F_in][F_out] row-major
    const float* __restrict__ bias, // [F_out]
    float* __restrict__ out, int F_in, int F_out, int m_tiles) {

    const int lane   = threadIdx.x;
    const int half   = lane >> 4;   // 0 or 1
    const int l16    = lane & 15;
    const int n_base = blockIdx.y * 16;

    int mt0 = blockIdx.x * MT;                    // first M tile of this wave
    if (mt0 + MT > m_tiles) mt0 = m_tiles - MT;   // overlap-recompute tail (requires m_tiles >= MT)

    v8f c[MT];
#pragma unroll
    for (int t = 0; t < MT; ++t) c[t] = (v8f){};

    const float* Ts[3] = {T0, T1, T2};

#pragma unroll
    for (int k3 = 0; k3 < 3; ++k3) {
        const float* A  = Ts[k3] + ((size_t)mt0 * 16 + l16) * F_in;
        const float* Wk = W + (size_t)k3 * F_in * F_out + n_base + l16;
        for (int k = 0; k < F_in; k += 4) {
            const float* bp = Wk + (size_t)(k + half * 2) * F_out;
            v2f b; b.x = bp[0]; b.y = bp[F_out];
#pragma unroll
            for (int t = 0; t < MT; ++t) {
                const float* ap = A + (size_t)t * 16 * F_in + k + half * 2;
                v2f a; a.x = ap[0]; a.y = ap[1];
                c[t] = __builtin_amdgcn_wmma_f32_16x16x4_f32(
                           false, a, false, b, (short)0, c[t], false, false);
            }
        }
    }

    const int col = n_base + l16;
    const float bv = bias[col];
#pragma unroll
    for (int t = 0; t < MT; ++t) {
        const int rb = (mt0 + t) * 16 + half * 8;
#pragma unroll
        for (int j = 0; j < 8; ++j) {
            float v = c[t][j] + bv;
            out[(size_t)(rb + j) * F_out + col] = v > 0.0f ? v : 0.0f;
        }
    }
}

// ---------------- pooling (batch ids are sorted) ----------------
__global__ __launch_bounds__(256) void k_pool(const float* __restrict__ h,
                                              const int* __restrict__ batch,
                                              float* __restrict__ sums,
                                              float* __restrict__ cnt, int n) {
    __shared__ float ls[NGRAPH * FHEAD];
    __shared__ float lc[NGRAPH];
    for (int i = threadIdx.x; i < NGRAPH * FHEAD; i += 256) ls[i] = 0.0f;
    for (int i = threadIdx.x; i < NGRAPH; i += 256) lc[i] = 0.0f;
    __syncthreads();

    const int chunk = 512;
    const int base  = blockIdx.x * chunk;

    for (int i = threadIdx.x; i < chunk * FHEAD; i += 256) {
        int node = base + i / FHEAD;
        int f    = i % FHEAD;
        if (node < n) {
            int g = batch[node];
            atomicAdd(&ls[g * FHEAD + f], h[(size_t)node * FHEAD + f]);
            if (f == 0) atomicAdd(&lc[g], 1.0f);
        }
    }
    __syncthreads();

    int last = base + chunk - 1; if (last >= n) last = n - 1;
    int g_lo = batch[base < n ? base : (n - 1)];
    int g_hi = batch[last];
    int span = (g_hi - g_lo + 1) * FHEAD;
    for (int i = threadIdx.x; i < span; i += 256) {
        int g = g_lo + i / FHEAD;
        int f = i % FHEAD;
        float v = ls[g * FHEAD + f];
        if (v != 0.0f) atomicAdd(&sums[g * FHEAD + f], v);
        if (f == 0 && lc[g] != 0.0f) atomicAdd(&cnt[g], lc[g]);
    }
}

// ---------------- head: out[g] = (sums[g]/max(cnt,1)) @ Wl + bl ----------------
__global__ __launch_bounds__(256) void k_head(const float* __restrict__ sums,
                                              const float* __restrict__ cnt,
                                              const float* __restrict__ Wl,
                                              const float* __restrict__ bl,
                                              float* __restrict__ out) {
    int t = threadIdx.x;           // 256 = 128 graphs * 2 outputs
    int g = t >> 1, j = t & 1;
    float c = cnt[g]; c = c > 1.0f ? c : 1.0f;
    float acc = bl[j];
#pragma unroll
    for (int f = 0; f < FHEAD; ++f)
        acc += (sums[g * FHEAD + f] / c) * Wl[f * 2 + j];
    out[g * 2 + j] = acc;
}

// ---------------- host side ----------------

static inline int cdiv(long long a, long long b) { return (int)((a + b - 1) / b); }

static void run_layer(const float* in, int Fin, int Fout,
                      const float* W, const float* b, float* out,
                      const float* w, const int* src, const int* dst,
                      float* t1, float* t2, int N, int E, hipStream_t stream) {
    size_t nf = (size_t)N * Fin;
    k_zero<<<cdiv(nf, 256), 256, 0, stream>>>(t1, nf);
    dim3 pgrid(cdiv(E, 256), Fin / 4);
    k_prop<<<pgrid, 256, 0, stream>>>(w, src, dst, in, t1, E, Fin, 1.0f);
    k_negcopy<<<cdiv(nf, 256), 256, 0, stream>>>(t2, in, nf);
    k_prop<<<pgrid, 256, 0, stream>>>(w, src, dst, t1, t2, E, Fin, 2.0f);
    int m_tiles = N / 16;
    dim3 ggrid(cdiv(m_tiles, MT), Fout / 16);
    k_cheb_gemm3<<<ggrid, 32, 0, stream>>>(in, t1, t2, W, b, out, Fin, Fout, m_tiles);
}

extern "C" void kernel_launch(void* const* d_in, const int* in_sizes, int n_in,
                              void* d_out, int out_size, void* d_ws, size_t ws_size,
                              hipStream_t stream) {
    const float* x    = (const float*)d_in[0];
    const int*   ei   = (const int*)d_in[1];
    const float* attr = (const float*)d_in[2];
    const int*   bat  = (const int*)d_in[3];
    const float* W1 = (const float*)d_in[4];  const float* b1 = (const float*)d_in[5];
    const float* W2 = (const float*)d_in[6];  const float* b2 = (const float*)d_in[7];
    const float* W3 = (const float*)d_in[8];  const float* b3 = (const float*)d_in[9];
    const float* Wl = (const float*)d_in[10]; const float* bl = (const float*)d_in[11];

    const int N = in_sizes[3];         // 50000
    const int E = in_sizes[2];         // 500000
    const int* src = ei;
    const int* dst = ei + E;

    // workspace carve-up (floats)
    float* ws = (float*)d_ws;
    float* deg  = ws;                       ws += N;              // also dinv (in place)
    float* w    = ws;                       ws += E;
    float* t1   = ws;                       ws += (size_t)N * 160;
    float* t2   = ws;                       ws += (size_t)N * 160;
    float* hA   = ws;                       ws += (size_t)N * 128;
    float* hB   = ws;                       ws += (size_t)N * 64;
    float* hC   = ws;                       ws += (size_t)N * 32;
    float* sums = ws;                       ws += NGRAPH * FHEAD;
    float* cnt  = ws;                       ws += NGRAPH;

    // normalization weights
    k_zero<<<cdiv(N, 256), 256, 0, stream>>>(deg, (size_t)N);
    k_degree<<<cdiv(E, 256), 256, 0, stream>>>(src, attr, deg, E);
    k_dinv<<<cdiv(N, 256), 256, 0, stream>>>(deg, N);
    k_edgew<<<cdiv(E, 256), 256, 0, stream>>>(src, dst, attr, deg, w, E);

    // three Chebyshev conv layers (WMMA GEMMs)
    run_layer(x,  160, 128, W1, b1, hA, w, src, dst, t1, t2, N, E, stream);
    run_layer(hA, 128,  64, W2, b2, hB, w, src, dst, t1, t2, N, E, stream);
    run_layer(hB,  64,  32, W3, b3, hC, w, src, dst, t1, t2, N, E, stream);

    // mean pool + linear head
    k_zero<<<1, 256, 0, stream>>>(sums, (size_t)(NGRAPH * FHEAD));
    k_zero<<<1, 256, 0, stream>>>(cnt, (size_t)NGRAPH);
    k_pool<<<cdiv(N, 512), 256, 0, stream>>>(hC, bat, sums, cnt, N);
    k_head<<<1, 256, 0, stream>>>(sums, cnt, Wl, bl, (float*)d_out);
}